// Grid_80582176408182
// MI455X (gfx1250) — compile-verified
//
#include <hip/hip_runtime.h>
#include <stdint.h>

#define NPTS   262144      // 32*8192
#define NPB    8192        // points per batch
#define NBATCH 32
#define NGRID  20
#define NCENT  8000        // 20^3
#define NSEL   800         // 10%
#define NGROUP 128
#define GSIZE  32
#define CHUNK  512
#define OUT_CB_OFF 393216  // 32*128*32*3

typedef __attribute__((ext_vector_type(2))) float v2f;
typedef __attribute__((ext_vector_type(8))) float v8f;
typedef __attribute__((ext_vector_type(4))) unsigned int u32x4;
typedef __attribute__((ext_vector_type(8))) int i32x8;
typedef __attribute__((ext_vector_type(4))) int i32x4;

// ---------------- Stage A: global stats (deterministic tree reduction) ----------------

__device__ __forceinline__ void combine10(float* a, const float* b) {
  a[0] = fminf(a[0], b[0]); a[1] = fminf(a[1], b[1]); a[2] = fminf(a[2], b[2]);
  a[3] = fmaxf(a[3], b[3]); a[4] = fmaxf(a[4], b[4]); a[5] = fmaxf(a[5], b[5]);
  a[6] += b[6]; a[7] += b[7]; a[8] += b[8]; a[9] += b[9];
}

__global__ __launch_bounds__(256) void k_reduce1(const float* __restrict__ xyz,
                                                 float* __restrict__ part) {
  __shared__ float s[256][10];
  const int tid = threadIdx.x;
  float acc[10] = {INFINITY, INFINITY, INFINITY, -INFINITY, -INFINITY, -INFINITY,
                   0.f, 0.f, 0.f, 0.f};
  const int base = blockIdx.x * 1024;
  for (int k = 0; k < 4; ++k) {
    int p = base + tid + k * 256;
    float x = xyz[3*p+0], y = xyz[3*p+1], z = xyz[3*p+2];
    acc[0] = fminf(acc[0], x); acc[1] = fminf(acc[1], y); acc[2] = fminf(acc[2], z);
    acc[3] = fmaxf(acc[3], x); acc[4] = fmaxf(acc[4], y); acc[5] = fmaxf(acc[5], z);
    acc[6] += x; acc[7] += y; acc[8] += z; acc[9] += x*x + y*y + z*z;
  }
  for (int i = 0; i < 10; ++i) s[tid][i] = acc[i];
  __syncthreads();
  for (int off = 128; off > 0; off >>= 1) {
    if (tid < off) combine10(s[tid], s[tid + off]);
    __syncthreads();
  }
  if (tid == 0)
    for (int i = 0; i < 10; ++i) part[blockIdx.x * 10 + i] = s[0][i];
}

__global__ __launch_bounds__(256) void k_reduce2(const float* __restrict__ part,
                                                 float* __restrict__ stats) {
  __shared__ float s[256][10];
  const int tid = threadIdx.x;
  for (int i = 0; i < 10; ++i) s[tid][i] = part[tid * 10 + i];
  __syncthreads();
  for (int off = 128; off > 0; off >>= 1) {
    if (tid < off) combine10(s[tid], s[tid + off]);
    __syncthreads();
  }
  if (tid == 0)
    for (int i = 0; i < 10; ++i) stats[i] = s[0][i];
}

// ---------------- Stage B1: tot per grid center ----------------

__global__ __launch_bounds__(256) void k_tot(const float* __restrict__ st,
                                             float* __restrict__ tot) {
  int c = blockIdx.x * 256 + threadIdx.x;
  if (c >= NCENT) return;
  float mnx = st[0], mny = st[1], mnz = st[2];
  float mxx = st[3], mxy = st[4], mxz = st[5];
  float Sx = st[6], Sy = st[7], Sz = st[8], Ssq = st[9];
  float lox = mnx + (mxx - mnx) / 21.f, hix = mxx - (mxx - mnx) / 21.f;
  float loy = mny + (mxy - mny) / 21.f, hiy = mxy - (mxy - mny) / 21.f;
  float loz = mnz + (mxz - mnz) / 21.f, hiz = mxz - (mxz - mnz) / 21.f;
  int i = c / 400, j = (c / 20) % 20, k = c % 20;
  float cx = lox + (hix - lox) * (float)i / 19.f;
  float cy = loy + (hiy - loy) * (float)j / 19.f;
  float cz = loz + (hiz - loz) * (float)k / 19.f;
  tot[c] = Ssq - 2.f * (cx * Sx + cy * Sy + cz * Sz)
         + (float)NPTS * (cx * cx + cy * cy + cz * cz);
}

// ---------------- Stage B2: select 800 smallest tot (ascending, stable) ----------------

__global__ __launch_bounds__(256) void k_select(const float* __restrict__ tot,
                                                const float* __restrict__ st,
                                                float* __restrict__ sel) {
  __shared__ float s_tot[NCENT];
  __shared__ unsigned long long s_key[256];
  const int tid = threadIdx.x;
  for (int i = tid; i < NCENT; i += 256) s_tot[i] = tot[i];
  float mnx = st[0], mny = st[1], mnz = st[2];
  float mxx = st[3], mxy = st[4], mxz = st[5];
  float lox = mnx + (mxx - mnx) / 21.f, hix = mxx - (mxx - mnx) / 21.f;
  float loy = mny + (mxy - mny) / 21.f, hiy = mxy - (mxy - mny) / 21.f;
  float loz = mnz + (mxz - mnz) / 21.f, hiz = mxz - (mxz - mnz) / 21.f;
  __syncthreads();
  for (int s = 0; s < NSEL; ++s) {
    unsigned long long best = ~0ull;
    for (int i = tid; i < NCENT; i += 256) {
      unsigned long long key =
          ((unsigned long long)__float_as_uint(s_tot[i]) << 32) | (unsigned)i;
      if (key < best) best = key;
    }
    s_key[tid] = best;
    __syncthreads();
    for (int off = 128; off > 0; off >>= 1) {
      if (tid < off && s_key[tid + off] < s_key[tid]) s_key[tid] = s_key[tid + off];
      __syncthreads();
    }
    if (tid == 0) {
      int c = (int)(s_key[0] & 0xFFFFFFFFu);
      int i = c / 400, j = (c / 20) % 20, kk = c % 20;
      sel[3 * s + 0] = lox + (hix - lox) * (float)i / 19.f;
      sel[3 * s + 1] = loy + (hiy - loy) * (float)j / 19.f;
      sel[3 * s + 2] = loz + (hiz - loz) * (float)kk / 19.f;
      s_tot[c] = INFINITY;
    }
    __syncthreads();
  }
}

// ---------------- Stage B3: farthest point sampling (sequential) ----------------

__global__ __launch_bounds__(256) void k_fps(const float* __restrict__ sel,
                                             float* __restrict__ ctr) {
  __shared__ float s_p[NSEL * 3];
  __shared__ float s_d[NSEL];
  __shared__ unsigned long long s_key[256];
  __shared__ int s_cur;
  const int tid = threadIdx.x;
  for (int i = tid; i < NSEL * 3; i += 256) s_p[i] = sel[i];
  for (int i = tid; i < NSEL; i += 256) s_d[i] = INFINITY;
  if (tid == 0) {
    s_cur = 0;
    ctr[0] = sel[0]; ctr[1] = sel[1]; ctr[2] = sel[2];
  }
  __syncthreads();
  for (int s = 1; s < NGROUP; ++s) {
    int cur = s_cur;
    float lx = s_p[3*cur], ly = s_p[3*cur+1], lz = s_p[3*cur+2];
    unsigned long long best = 0ull;
    for (int i = tid; i < NSEL; i += 256) {
      float dx = s_p[3*i] - lx, dy = s_p[3*i+1] - ly, dz = s_p[3*i+2] - lz;
      float d = fminf(s_d[i], dx*dx + dy*dy + dz*dz);
      s_d[i] = d;
      unsigned long long key = ((unsigned long long)__float_as_uint(d) << 32)
                             | (unsigned)(0xFFFFFFFFu - (unsigned)i);
      if (key > best) best = key;
    }
    s_key[tid] = best;
    __syncthreads();
    for (int off = 128; off > 0; off >>= 1) {
      if (tid < off && s_key[tid + off] > s_key[tid]) s_key[tid] = s_key[tid + off];
      __syncthreads();
    }
    if (tid == 0) {
      int nxt = (int)(0xFFFFFFFFu - (unsigned)(s_key[0] & 0xFFFFFFFFu));
      s_cur = nxt;
      ctr[3*s+0] = s_p[3*nxt+0]; ctr[3*s+1] = s_p[3*nxt+1]; ctr[3*s+2] = s_p[3*nxt+2];
    }
    __syncthreads();
  }
}

// ---------------- TDM: stage one 512-point chunk (1536 f32) into LDS ----------------
// D# per CDNA5 ISA §8.3–8.5: group0={count=1, lds_addr, global_addr, type=2};
// group1={data_size=4B, tensor_dim0=tile_dim0=nElems, 1-D tile}; groups 2/3/4 zero.
// This toolchain exposes the 6-arg builtin form (extra int32x8 group + cpol).

__device__ __forceinline__ void tdm_load_chunk(const float* gsrc, unsigned ldsOff,
                                               int nElems) {
  unsigned long long ga = (unsigned long long)(uintptr_t)gsrc;
  u32x4 g0;
  g0[0] = 1u;                                            // count=1, user mode
  g0[1] = ldsOff;                                        // lds_addr (bytes)
  g0[2] = (unsigned)(ga & 0xFFFFFFFFu);                  // global_addr[31:0]
  g0[3] = (unsigned)((ga >> 32) & 0x01FFFFFFu) | 0x80000000u; // ga[56:32] | type=2
  i32x8 g1;
  g1[0] = (int)(2u << 16);                               // data_size=4B, mask=0
  g1[1] = (int)(((unsigned)nElems & 0xFFFFu) << 16);     // tensor_dim0[15:0]
  g1[2] = (int)((unsigned)nElems >> 16);                 // tensor_dim0[31:16], dim1=0
  g1[3] = (int)(((unsigned)nElems & 0xFFFFu) << 16);     // tile_dim0
  g1[4] = 0;                                             // tile_dim1=0, tile_dim2=0
  g1[5] = nElems;                                        // stride0 (unused for 1-D)
  g1[6] = 0;
  g1[7] = 0;
  i32x4 z4 = {0, 0, 0, 0};
  i32x8 z8 = {0, 0, 0, 0, 0, 0, 0, 0};
  __builtin_amdgcn_tensor_load_to_lds(g0, g1, z4, z4, z8, 0);
}

// ---------------- wave argmax of (bd, bi): T = max, ml = owning lane ----------------

__device__ __forceinline__ void waveArgMax(float bd, unsigned bi, float& T, int& ml) {
  unsigned long long key =
      ((unsigned long long)__float_as_uint(bd) << 32) | (unsigned long long)bi;
  int l = (int)(threadIdx.x & 31);
#pragma unroll
  for (int off = 16; off > 0; off >>= 1) {
    unsigned long long ok = __shfl_xor(key, off, 32);
    int ol = __shfl_xor(l, off, 32);
    if (ok > key) { key = ok; l = ol; }
  }
  T = __uint_as_float((unsigned)(key >> 32));
  ml = l;
}

// ---------------- Stage C: TDM staging + WMMA distances + top-32 + gather ----------------
// One 512-thread block per (batch, tile of 16 groups). 16 waves.
// Distance via V_WMMA_F32_16X16X4_F32: A=(-2cx,-2cy,-2cz,||c||^2), B=(x,y,z,1), C=||x||^2.

__global__ __launch_bounds__(512) void k_group(const float* __restrict__ xyz,
                                               const float* __restrict__ ctr,
                                               float* __restrict__ out) {
  const int b    = blockIdx.y;
  const int gt   = blockIdx.x;          // group tile 0..7
  const int tid  = threadIdx.x;
  const int wave = tid >> 5;            // 0..15
  const int lane = tid & 31;

  __shared__ float s_ctr[16][4];
  __shared__ float s_pts[2][CHUNK * 3];  // double-buffered TDM staging
  __shared__ float s_dist[16][CHUNK];
  __shared__ unsigned long long s_sort[16][32];

  if (tid < 16) {
    float cx = ctr[(gt * 16 + tid) * 3 + 0];
    float cy = ctr[(gt * 16 + tid) * 3 + 1];
    float cz = ctr[(gt * 16 + tid) * 3 + 2];
    s_ctr[tid][0] = cx; s_ctr[tid][1] = cy; s_ctr[tid][2] = cz;
    s_ctr[tid][3] = cx*cx + cy*cy + cz*cz;
  }

  const float* xb = xyz + (size_t)b * NPB * 3;
  const unsigned ldsPts0 = (unsigned)(uintptr_t)(void*)&s_pts[0][0]; // flat low32 = LDS offset
  const unsigned ldsPts1 = (unsigned)(uintptr_t)(void*)&s_pts[1][0];

  // kick off chunk 0 DMA (wave 0 only; TDM ignores EXEC, tracked by TENSORcnt)
  if (wave == 0) tdm_load_chunk(xb, ldsPts0, CHUNK * 3);
  __syncthreads();

  // A operand (constant). lane<16 -> (K0,K1)=(-2cx,-2cy); else (K2,K3)=(-2cz,||c||^2)
  const int gm = lane & 15;
  float a0 = (lane < 16) ? -2.f * s_ctr[gm][0] : -2.f * s_ctr[gm][2];
  float a1 = (lane < 16) ? -2.f * s_ctr[gm][1] : s_ctr[gm][3];
  v2f av; av[0] = a0; av[1] = a1;

  float    bd = INFINITY;   // distributed top-32: one slot per lane
  unsigned bi = 0u;
  float T; int ml;          // lazy threshold: max of bd and its owner lane
  waveArgMax(bd, bi, T, ml);

  for (int ch = 0; ch < 16; ++ch) {
    const int cbase = ch * CHUNK;

    // prefetch next chunk; wait until current chunk's DMA (older) is complete
    if (wave == 0) {
      if (ch < 15) {
        tdm_load_chunk(xb + 3 * (cbase + CHUNK), ((ch + 1) & 1) ? ldsPts1 : ldsPts0,
                       CHUNK * 3);
        __builtin_amdgcn_s_wait_tensorcnt(1);
      } else {
        __builtin_amdgcn_s_wait_tensorcnt(0);
      }
    }
    __syncthreads();  // (A) current chunk's points visible to all waves

    const float* pb = &s_pts[ch & 1][0];

    // ---- WMMA phase: 2 tiles per wave (uniform control flow, EXEC all-ones) ----
    for (int t2 = 0; t2 < 2; ++t2) {
      int tile = wave * 2 + t2;               // 0..31
      int pl   = tile * 16 + (lane & 15);     // chunk-local point index
      float px = pb[3 * pl + 0];
      float py = pb[3 * pl + 1];
      float pz = pb[3 * pl + 2];
      float sn = px*px + py*py + pz*pz;
      float b0 = (lane < 16) ? px : pz;       // B: lane<16 -> (K0,K1)=(x,y); else (K2,K3)=(z,1)
      float b1 = (lane < 16) ? py : 1.0f;
      v2f bv; bv[0] = b0; bv[1] = b1;
      v8f cv = { sn, sn, sn, sn, sn, sn, sn, sn };
      v8f dv = __builtin_amdgcn_wmma_f32_16x16x4_f32(
          false, av, false, bv, (short)0, cv, false, false);
      int mrow = (lane >> 4) * 8;
#pragma unroll
      for (int r = 0; r < 8; ++r) s_dist[mrow + r][pl] = dv[r];
    }
    __syncthreads();  // (B) distances ready

    // ---- selection phase: wave w scans group row w; lazy threshold ----
#pragma unroll 1
    for (int i = 0; i < 16; ++i) {
      int j = lane + 32 * i;
      float    cd = s_dist[wave][j];
      unsigned ci = (unsigned)(cbase + j);
      bool pend = true;
      unsigned long long mask = __ballot(cd < T);
      while (mask) {
        int src = __ffsll(mask) - 1;          // lowest lane = lowest index first
        float    dN = __shfl(cd, src, 32);
        unsigned iN = (unsigned)__shfl((int)ci, src, 32);
        if (lane == src) pend = false;
        if (lane == ml) { bd = dN; bi = iN; } // evict current max slot
        waveArgMax(bd, bi, T, ml);            // refresh threshold after insertion
        mask = __ballot(pend && (cd < T));
      }
    }
    __syncthreads();  // (C) selection done; s_pts/s_dist reusable next chunk
  }

  // ---- sort the 32 winners ascending by (d, idx) -> top_k output order ----
  unsigned long long myKey =
      ((unsigned long long)__float_as_uint(bd) << 32) | (unsigned long long)bi;
  int rank = 0;
#pragma unroll 1
  for (int j = 0; j < 32; ++j) {
    unsigned long long kj = __shfl(myKey, j, 32);
    rank += (kj < myKey) ? 1 : 0;
  }
  s_sort[wave][rank] = myKey;
  __syncthreads();
  unsigned pi = (unsigned)(s_sort[wave][lane] & 0xFFFFFFFFu);

  const int g = gt * 16 + wave;
  float cx = s_ctr[wave][0], cy = s_ctr[wave][1], cz = s_ctr[wave][2];
  size_t ob = (((size_t)b * NGROUP + g) * GSIZE + lane) * 3;
  out[ob + 0] = xb[3 * pi + 0] - cx;
  out[ob + 1] = xb[3 * pi + 1] - cy;
  out[ob + 2] = xb[3 * pi + 2] - cz;
  if (lane < 3) out[OUT_CB_OFF + ((size_t)b * NGROUP + g) * 3 + lane] = s_ctr[wave][lane];
}

// ---------------- launch ----------------

extern "C" void kernel_launch(void* const* d_in, const int* in_sizes, int n_in,
                              void* d_out, int out_size, void* d_ws, size_t ws_size,
                              hipStream_t stream) {
  (void)in_sizes; (void)n_in; (void)out_size; (void)ws_size;
  const float* xyz = (const float*)d_in[0];
  float* out = (float*)d_out;
  float* ws  = (float*)d_ws;

  float* part  = ws;            // 256*10 = 2560
  float* stats = ws + 2560;     // 10 (pad to 16)
  float* tot   = ws + 2576;     // 8000
  float* sel   = ws + 10576;    // 800*3 = 2400
  float* ctrs  = ws + 12976;    // 128*3 = 384   (total ~53 KB)

  k_reduce1<<<256, 256, 0, stream>>>(xyz, part);
  k_reduce2<<<1, 256, 0, stream>>>(part, stats);
  k_tot<<<(NCENT + 255) / 256, 256, 0, stream>>>(stats, tot);
  k_select<<<1, 256, 0, stream>>>(tot, stats, sel);
  k_fps<<<1, 256, 0, stream>>>(sel, ctrs);
  k_group<<<dim3(8, NBATCH), 512, 0, stream>>>(xyz, ctrs, out);
}